// Aggregation_Separation_Loss_14534169330359
// MI455X (gfx1250) — compile-verified
//
#include <hip/hip_runtime.h>
#include <hip/hip_bf16.h>
#include <math.h>

// Problem constants (match reference setup_inputs: N=768, D=256, labels in [0,10))
#define NROW   768
#define DDIM   256
#define NT     48          // NROW / 16 tiles per side
#define LDSTR  260         // padded LDS row stride (floats): bank-conflict-free
#define CONSTRAINT 10.0f

typedef float v2f __attribute__((ext_vector_type(2)));
typedef float v8f __attribute__((ext_vector_type(8)));

// ---------------------------------------------------------------------------
// Kernel 1: row squared norms  Q[i] = sum_d x[i][d]^2
// ---------------------------------------------------------------------------
__global__ __launch_bounds__(256) void row_norms_kernel(
    const float* __restrict__ X, float* __restrict__ Q) {
  __shared__ float red[256];
  const int row = blockIdx.x;
  const float v = X[row * DDIM + threadIdx.x];
  red[threadIdx.x] = v * v;
  __syncthreads();
  for (int s = 128; s > 0; s >>= 1) {
    if (threadIdx.x < s) red[threadIdx.x] += red[threadIdx.x + s];
    __syncthreads();
  }
  if (threadIdx.x == 0) Q[row] = red[0];
}

// ---------------------------------------------------------------------------
// Kernel 2: one wave32 per 16x16 tile of the (i,j) pair matrix.
//
// Exact identity (beta = 1):
//   huber(a) = 0.5*a^2 - 0.5*max(|a|-1, 0)^2
// Summed over d, with a = x_i[d] - x_j[d]:
//   sum_d 0.5*a^2 = 0.5*Q_i + 0.5*Q_j - <x_i, x_j>      <- Gram via WMMA f32
//   correction    = 0.5 * sum_d max(|a|-1,0)^2          <- elementwise VALU
//
// Gram and correction are fused per 4-wide K chunk; the Gram accumulator is
// split 4 ways to break the WMMA->WMMA RAW chain (hazard table 7.12.1), so
// the scheduler can overlap WMMA latency with the correction VALU stream.
// ---------------------------------------------------------------------------
__global__ __launch_bounds__(32) void tile_kernel(
    const float* __restrict__ X, const int* __restrict__ labels,
    const float* __restrict__ Q, float* __restrict__ partials) {
  __shared__ float As[16 * LDSTR];
  __shared__ float Bs[16 * LDSTR];
  __shared__ float red[64];

  const int iT = blockIdx.y, jT = blockIdx.x;
  const int tileId = iT * NT + jT;
  const int lane = threadIdx.x;

  // Symmetry: sl1 and the label mask are symmetric; compute upper triangle
  // only, scale off-diagonal tiles by 2. Lower tiles just zero their slot.
  if (jT < iT) {
    if (lane == 0) { partials[2 * tileId] = 0.0f; partials[2 * tileId + 1] = 0.0f; }
    return;
  }

  const int i0 = iT * 16, j0 = jT * 16;

  // ---- Stage A (rows i0..i0+15) and B (rows j0..j0+15) panels into LDS ----
  // 16 rows x 64 float4 per panel; 32 lanes -> coalesced b128 global loads.
  const float4* __restrict__ Xv = reinterpret_cast<const float4*>(X);
  for (int t = lane; t < 16 * (DDIM / 4); t += 32) {
    const int row = t >> 6;          // 0..15
    const int c4  = t & 63;          // float4 column 0..63
    float4 va = Xv[(size_t)(i0 + row) * (DDIM / 4) + c4];
    float4 vb = Xv[(size_t)(j0 + row) * (DDIM / 4) + c4];
    *reinterpret_cast<float4*>(&As[row * LDSTR + c4 * 4]) = va;
    *reinterpret_cast<float4*>(&Bs[row * LDSTR + c4 * 4]) = vb;
  }
  __syncthreads();

  const int half = lane >> 4;        // 0: lanes 0-15, 1: lanes 16-31
  const int n    = lane & 15;        // fragment row index (M for A, N for B)

  // ISA 32-bit A 16x4 layout: lane = M + 16*(K>>1), vgpr = K&1  -> per lane
  // a float2 at k = 4c + 2*half. B (4x16) mirrors with lane = N + 16*(K>>1).
  // C-fragment cell mapping: lane-half 0 -> M = r,   N = n
  //                          lane-half 1 -> M = 8+r, N = n
  const int mbase = 8 * half;

  v8f acc[4];
  acc[0] = (v8f){}; acc[1] = (v8f){}; acc[2] = (v8f){}; acc[3] = (v8f){};
  float corr[8] = {0.f, 0.f, 0.f, 0.f, 0.f, 0.f, 0.f, 0.f};

#pragma unroll 4
  for (int c = 0; c < DDIM / 4; ++c) {
    const int k4 = 4 * c;

    // --- Gram chunk: D += A(16x4) * B(4x16), round-robin accumulator ---
    v2f a = *reinterpret_cast<const v2f*>(&As[n * LDSTR + k4 + 2 * half]);
    v2f b = *reinterpret_cast<const v2f*>(&Bs[n * LDSTR + k4 + 2 * half]);
    acc[c & 3] = __builtin_amdgcn_wmma_f32_16x16x4_f32(
        /*neg_a=*/false, a, /*neg_b=*/false, b,
        /*c_mod=*/(short)0, acc[c & 3], /*reuse_a=*/false, /*reuse_b=*/false);

    // --- Correction chunk: corr[r] += sum over 4 d of max(|a|-1,0)^2 ---
    // b row: lanes n and 16+n read identical addresses -> LDS broadcast
    float4 bv = *reinterpret_cast<const float4*>(&Bs[n * LDSTR + k4]);
#pragma unroll
    for (int r = 0; r < 8; ++r) {
      // a row: uniform address across each half-wave -> LDS broadcast
      float4 av = *reinterpret_cast<const float4*>(&As[(mbase + r) * LDSTR + k4]);
      float t0 = fmaxf(fabsf(av.x - bv.x) - 1.0f, 0.0f);
      float t1 = fmaxf(fabsf(av.y - bv.y) - 1.0f, 0.0f);
      float t2 = fmaxf(fabsf(av.z - bv.z) - 1.0f, 0.0f);
      float t3 = fmaxf(fabsf(av.w - bv.w) - 1.0f, 0.0f);
      float s = corr[r];
      s = fmaf(t0, t0, s);
      s = fmaf(t1, t1, s);
      s = fmaf(t2, t2, s);
      s = fmaf(t3, t3, s);
      corr[r] = s;
    }
  }

  // Combine the 4 round-robin accumulators.
  v8f accT = (acc[0] + acc[1]) + (acc[2] + acc[3]);

  // ---- Epilogue: assemble sl1, mask by labels, reduce to tile partials ----
  const int   lj = labels[j0 + n];
  const float qj = Q[j0 + n];
  float innerP = 0.0f, outerP = 0.0f;
#pragma unroll
  for (int r = 0; r < 8; ++r) {
    const int i = i0 + mbase + r;
    const float hub = 0.5f * (Q[i] + qj) - accT[r] - 0.5f * corr[r];
    const float sl1 = hub * (1.0f / DDIM);
    const bool same = (labels[i] == lj);
    innerP += same ? sl1 : 0.0f;
    outerP += same ? 0.0f : sl1;
  }

  red[lane]      = innerP;
  red[32 + lane] = outerP;
  __syncthreads();
  if (lane == 0) {
    const float scale = (iT == jT) ? 1.0f : 2.0f;  // symmetry doubling
    float si = 0.0f, so = 0.0f;
    for (int l = 0; l < 32; ++l) { si += red[l]; so += red[32 + l]; }
    partials[2 * tileId]     = si * scale;
    partials[2 * tileId + 1] = so * scale;
  }
}

// ---------------------------------------------------------------------------
// Kernel 3: deterministic final reduction (single block, fixed order).
// ---------------------------------------------------------------------------
__global__ __launch_bounds__(256) void finalize_kernel(
    const float* __restrict__ partials, const float* __restrict__ Q,
    const int* __restrict__ labels, float* __restrict__ out) {
  __shared__ double dred[256];
  __shared__ long long cred[256];
  __shared__ int slab[NROW];

  const int tid = threadIdx.x;
  for (int t = tid; t < NROW; t += 256) slab[t] = labels[t];
  __syncthreads();

  // Sum tile partials (fixed per-thread stride order -> deterministic).
  double si = 0.0, so = 0.0;
  for (int t = tid; t < NT * NT; t += 256) {
    si += (double)partials[2 * t];
    so += (double)partials[2 * t + 1];
  }

  // Count same-label pairs over the full NxN matrix.
  long long cnt = 0;
  for (int i = tid; i < NROW; i += 256) {
    const int li = slab[i];
    for (int j = 0; j < NROW; ++j) cnt += (li == slab[j]) ? 1 : 0;
  }

  // Norm penalty: mean((||x_i|| - 10)^2)
  double pp = 0.0;
  for (int i = tid; i < NROW; i += 256) {
    const double d = (double)sqrtf(Q[i]) - (double)CONSTRAINT;
    pp += d * d;
  }

  // Fixed-tree reductions.
  dred[tid] = si; __syncthreads();
  for (int s = 128; s > 0; s >>= 1) { if (tid < s) dred[tid] += dred[tid + s]; __syncthreads(); }
  const double innerT = dred[0]; __syncthreads();

  dred[tid] = so; __syncthreads();
  for (int s = 128; s > 0; s >>= 1) { if (tid < s) dred[tid] += dred[tid + s]; __syncthreads(); }
  const double outerT = dred[0]; __syncthreads();

  dred[tid] = pp; __syncthreads();
  for (int s = 128; s > 0; s >>= 1) { if (tid < s) dred[tid] += dred[tid + s]; __syncthreads(); }
  const double penT = dred[0]; __syncthreads();

  cred[tid] = cnt; __syncthreads();
  for (int s = 128; s > 0; s >>= 1) { if (tid < s) cred[tid] += cred[tid + s]; __syncthreads(); }

  if (tid == 0) {
    const long long n1 = cred[0];
    const long long n2 = (long long)NROW * NROW - n1;
    const double loss_inner = (n1 > 0) ? innerT / (double)n1 : innerT;
    const double loss_outer = (n2 > 0) ? outerT / (double)n2 : outerT;
    out[0] = (float)loss_inner;
    out[1] = (float)loss_outer;
    out[2] = (float)(penT / (double)NROW);
  }
}

// ---------------------------------------------------------------------------
extern "C" void kernel_launch(void* const* d_in, const int* in_sizes, int n_in,
                              void* d_out, int out_size, void* d_ws, size_t ws_size,
                              hipStream_t stream) {
  (void)in_sizes; (void)n_in; (void)out_size; (void)ws_size;
  const float* X      = (const float*)d_in[0];   // [768, 256] fp32
  const int*   labels = (const int*)d_in[1];     // [768] int32
  float*       out    = (float*)d_out;           // 3 floats

  // Workspace layout: Q[768] | partials[2 * 48 * 48]  (21.5 KB)
  float* Q        = (float*)d_ws;
  float* partials = Q + NROW;

  row_norms_kernel<<<NROW, 256, 0, stream>>>(X, Q);

  dim3 grid(NT, NT);
  tile_kernel<<<grid, 32, 0, stream>>>(X, labels, Q, partials);

  finalize_kernel<<<1, 256, 0, stream>>>(partials, Q, labels, out);
}